// QuantumClassifier_78743930405220
// MI455X (gfx1250) — compile-verified
//
#include <hip/hip_runtime.h>
#include <hip/hip_bf16.h>

typedef __attribute__((ext_vector_type(2))) float v2f;
typedef __attribute__((ext_vector_type(8))) float v8f;

#define NQ 4
#define DIM 16   // 2^NQ amplitudes

// ---------------------------------------------------------------------------
// Kernel 1: build M = Re(U^H Z0 U) from the shared variational weights.
// U is the fixed 16x16 unitary of the 2 variational layers (RY,RZ per qubit +
// CNOT ring), exactly matching the reference gate order. Threads 0..15 each
// evolve one basis column; then all 256 threads form one M entry each.
// Wire i maps to bit (3-i) of the flat index (wire 0 = MSB, per the reshape).
// ---------------------------------------------------------------------------
__global__ __launch_bounds__(256) void qc_build_M(const float* __restrict__ w,
                                                  float* __restrict__ M) {
    __shared__ float Ur[DIM][DIM];  // [n][col]
    __shared__ float Ui[DIM][DIM];
    const int t = threadIdx.x;

    if (t < DIM) {
        float ar[DIM], ai[DIM];
        #pragma unroll
        for (int n = 0; n < DIM; ++n) { ar[n] = (n == t) ? 1.0f : 0.0f; ai[n] = 0.0f; }

        #pragma unroll
        for (int layer = 0; layer < 2; ++layer) {
            #pragma unroll
            for (int q = 0; q < NQ; ++q) {
                const int bit = 1 << (3 - q);
                // RY(w[layer][q][0])
                {
                    const float th = 0.5f * w[(layer * NQ + q) * 2 + 0];
                    const float c = cosf(th), s = sinf(th);
                    #pragma unroll
                    for (int n = 0; n < DIM; ++n) {
                        if (!(n & bit)) {
                            const int n1 = n | bit;
                            const float r0 = ar[n], i0 = ai[n];
                            const float r1 = ar[n1], i1 = ai[n1];
                            ar[n]  = c * r0 - s * r1;  ai[n]  = c * i0 - s * i1;
                            ar[n1] = s * r0 + c * r1;  ai[n1] = s * i0 + c * i1;
                        }
                    }
                }
                // RZ(w[layer][q][1]): amp *= exp(-+ i th/2) by target bit
                {
                    const float ph = 0.5f * w[(layer * NQ + q) * 2 + 1];
                    const float cp = cosf(ph), sp = sinf(ph);
                    #pragma unroll
                    for (int n = 0; n < DIM; ++n) {
                        const float sg = (n & bit) ? sp : -sp;  // exp(+i)/exp(-i)
                        const float r = ar[n], i = ai[n];
                        ar[n] = cp * r - sg * i;
                        ai[n] = cp * i + sg * r;
                    }
                }
            }
            // CNOT ring: (0,1),(1,2),(2,3),(3,0): swap target pair where control=1
            const int ctrl[4] = {0, 1, 2, 3};
            const int targ[4] = {1, 2, 3, 0};
            #pragma unroll
            for (int g = 0; g < 4; ++g) {
                const int cb = 1 << (3 - ctrl[g]);
                const int tb = 1 << (3 - targ[g]);
                #pragma unroll
                for (int n = 0; n < DIM; ++n) {
                    if ((n & cb) && !(n & tb)) {
                        const int n1 = n | tb;
                        float tmp;
                        tmp = ar[n]; ar[n] = ar[n1]; ar[n1] = tmp;
                        tmp = ai[n]; ai[n] = ai[n1]; ai[n1] = tmp;
                    }
                }
            }
        }
        #pragma unroll
        for (int n = 0; n < DIM; ++n) { Ur[n][t] = ar[n]; Ui[n][t] = ai[n]; }
    }
    __syncthreads();

    // M[i][j] = sum_n z(n) * (Ur[n][i]Ur[n][j] + Ui[n][i]Ui[n][j]), z = +-1 by MSB
    const int i = t >> 4, j = t & 15;
    float acc = 0.0f;
    #pragma unroll
    for (int n = 0; n < DIM; ++n) {
        const float z = (n & 8) ? -1.0f : 1.0f;
        acc += z * (Ur[n][i] * Ur[n][j] + Ui[n][i] * Ui[n][j]);
    }
    M[t] = acc;
}

// ---------------------------------------------------------------------------
// Kernel 2: per wave, 16 samples. Build real encoded vectors v (products of
// half-angle cos/sin), compute C = V * M with 4 chained v_wmma_f32_16x16x4_f32
// (K = 16 split into 4), then out[m] = sum_p V[m][p] * C[m][p] via a per-wave
// LDS tile. 8 waves / block, 2048 blocks cover BATCH = 262144 exactly.
// ---------------------------------------------------------------------------
__global__ __launch_bounds__(256) void qc_eval(const float4* __restrict__ x4,
                                               const float* __restrict__ M,
                                               float* __restrict__ out) {
    const int lane   = threadIdx.x & 31;
    const int waveId = threadIdx.x >> 5;
    const int gwave  = blockIdx.x * 8 + waveId;
    const int col    = lane & 15;
    const bool hi    = lane >= 16;
    const int s      = gwave * 16 + col;   // both half-waves mirror the same sample

    // --- load x[s][0..3] and build v[16] = tensor product of (cos,sin) pairs ---
    const float4 xx = x4[s];
    const float c0 = __cosf(0.5f * xx.x), s0 = __sinf(0.5f * xx.x);
    const float c1 = __cosf(0.5f * xx.y), s1 = __sinf(0.5f * xx.y);
    const float c2 = __cosf(0.5f * xx.z), s2 = __sinf(0.5f * xx.z);
    const float c3 = __cosf(0.5f * xx.w), s3 = __sinf(0.5f * xx.w);

    const float p00 = c0 * c1, p01 = c0 * s1, p10 = s0 * c1, p11 = s0 * s1;
    const float q00 = c2 * c3, q01 = c2 * s3, q10 = s2 * c3, q11 = s2 * s3;

    float v[DIM];
    v[0]  = p00 * q00; v[1]  = p00 * q01; v[2]  = p00 * q10; v[3]  = p00 * q11;
    v[4]  = p01 * q00; v[5]  = p01 * q01; v[6]  = p01 * q10; v[7]  = p01 * q11;
    v[8]  = p10 * q00; v[9]  = p10 * q01; v[10] = p10 * q10; v[11] = p10 * q11;
    v[12] = p11 * q00; v[13] = p11 * q01; v[14] = p11 * q10; v[15] = p11 * q11;

    // --- C = V * M via 4 chained WMMA f32 16x16x4 ---
    // A (16x4): lanes 0-15 hold K=4k+0,4k+1; lanes 16-31 hold K=4k+2,4k+3 (row = lane%16)
    // B (4x16): VGPR r -> row 4k + r + (hi?2:0), col = lane%16
    v8f C = {};
    #pragma unroll
    for (int k = 0; k < 4; ++k) {
        v2f A;
        A.x = hi ? v[4 * k + 2] : v[4 * k + 0];
        A.y = hi ? v[4 * k + 3] : v[4 * k + 1];
        const int r0 = 4 * k + (hi ? 2 : 0);
        v2f B;
        B.x = M[r0 * DIM + col];
        B.y = M[(r0 + 1) * DIM + col];
        C = __builtin_amdgcn_wmma_f32_16x16x4_f32(
                /*neg_a=*/false, A, /*neg_b=*/false, B,
                /*c_mod=*/(short)0, C, /*reuse_a=*/false, /*reuse_b=*/false);
    }

    // --- out[m] = sum_p V[m][p] * C[m][p] via per-wave LDS tile ---
    // C layout: VGPR r -> (m = r + (hi?8:0), p = lane%16)
    __shared__ float tile[8][DIM][DIM + 1];
    const int mbase = hi ? 8 : 0;
    #pragma unroll
    for (int r = 0; r < 8; ++r)
        tile[waveId][mbase + r][col] = C[r];

    __syncthreads();

    if (lane < 16) {  // lane l owns sample l's v values
        float acc = 0.0f;
        #pragma unroll
        for (int p = 0; p < DIM; ++p)
            acc += v[p] * tile[waveId][lane][p];
        out[s] = acc;
    }
}

extern "C" void kernel_launch(void* const* d_in, const int* in_sizes, int n_in,
                              void* d_out, int out_size, void* d_ws, size_t ws_size,
                              hipStream_t stream) {
    const float* x = (const float*)d_in[0];      // [262144, 4] float32
    const float* w = (const float*)d_in[1];      // [2, 4, 2]  float32
    float* M   = (float*)d_ws;                   // 16x16 effective observable
    float* out = (float*)d_out;                  // [262144, 1] float32

    qc_build_M<<<1, 256, 0, stream>>>(w, M);

    const int batch = in_sizes[0] / 4;           // 262144
    const int waves = batch / 16;                // 16384 tiles of 16 samples
    const int blocks = waves / 8;                // 8 waves (256 thr) per block
    qc_eval<<<blocks, 256, 0, stream>>>((const float4*)x, M, out);
}